// GNNML1_64991445123376
// MI455X (gfx1250) — compile-verified
//
#include <hip/hip_runtime.h>

#define N_NODES 50000
#define N_EDGES 800000
#define N_GRAPHS 64
#define NOUT 32
#define NIN 96
#define NTILES (N_NODES / 16)   // 3125, exact
#define CAP 64                   // per-node in-edge bucket capacity (deg ~ Poisson(16))

typedef __attribute__((ext_vector_type(2))) float v2f;
typedef __attribute__((ext_vector_type(8))) float v8f;

// ---------------------------------------------------------------------------
// One-time per launch: bucket edges by destination node.
// cnt[d] = in-degree, eidx[d*CAP + slot] = src node of that in-edge.
// ---------------------------------------------------------------------------
__global__ void __launch_bounds__(256)
build_csr_kernel(const int* __restrict__ src, const int* __restrict__ dst,
                 int* __restrict__ cnt, int* __restrict__ eidx)
{
    int e = blockIdx.x * 256 + threadIdx.x;
    if (e >= N_EDGES) return;
    int d = dst[e];
    int slot = atomicAdd(&cnt[d], 1);
    if (slot < CAP) eidx[(size_t)d * CAP + slot] = src[e];
}

// ---------------------------------------------------------------------------
// Block 1: in=2.  K=2 GEMMs are cheaper as plain FMAs.
// Writes H[:,0:32] (h1), H[:,64:96] (h3), and pre = x @ conv1_W (no bias yet).
// ---------------------------------------------------------------------------
__global__ void __launch_bounds__(256)
block1_kernel(const float* __restrict__ x,
              const float* __restrict__ Wv,                               // conv1_W [2,32]
              const float* __restrict__ Wa, const float* __restrict__ Ba, // fc11
              const float* __restrict__ Wb, const float* __restrict__ Bb, // fc12
              const float* __restrict__ Wc, const float* __restrict__ Bc, // fc13
              float* __restrict__ H, float* __restrict__ pre)
{
    int idx = blockIdx.x * 256 + threadIdx.x;
    if (idx >= N_NODES * NOUT) return;
    int node = idx >> 5;
    int c    = idx & 31;
    float x0 = x[node * 2 + 0];
    float x1 = x[node * 2 + 1];
    float h1 = fmaxf(fmaf(x1, Wa[32 + c], fmaf(x0, Wa[c], Ba[c])), 0.0f);
    float vb = fmaf(x1, Wb[32 + c], fmaf(x0, Wb[c], Bb[c]));
    float vc = fmaf(x1, Wc[32 + c], fmaf(x0, Wc[c], Bc[c]));
    float h3 = fmaxf(vb * vc, 0.0f);
    pre[idx] = fmaf(x1, Wv[32 + c], x0 * Wv[c]);
    size_t r = (size_t)node * NIN;
    H[r + c]      = h1;
    H[r + 64 + c] = h3;
}

// ---------------------------------------------------------------------------
// Blocks 2..5 fused GEMM: one wave = one 16-row tile of H [50000,96].
// Computes H@fcA, H@fcB, H@fcC, H@convW with V_WMMA_F32_16X16X4_F32.
// 4 weight matrices (96x32 each, 48 KB total) staged in LDS.
// Epilogue: h1 = relu(.+Ba) -> H[:,0:32]; h3 = relu((.+Bb)*(.+Bc)) -> H[:,64:96];
// raw conv pre-activation -> pre[:,0:32].  In-place on H (per-row ownership).
// ---------------------------------------------------------------------------
__global__ void __launch_bounds__(256)
gemm96_kernel(const float* H,
              const float* __restrict__ Wa, const float* __restrict__ Ba,
              const float* __restrict__ Wb, const float* __restrict__ Bb,
              const float* __restrict__ Wc, const float* __restrict__ Bc,
              const float* __restrict__ Wv,
              float* Hn, float* __restrict__ pre, int nTiles)
{
    __shared__ float lw[4 * NIN * NOUT];   // 12288 floats = 48 KB: [A | B | C | conv]
    for (int i = threadIdx.x; i < NIN * NOUT; i += 256) {
        lw[i]         = Wa[i];
        lw[3072 + i]  = Wb[i];
        lw[6144 + i]  = Wc[i];
        lw[9216 + i]  = Wv[i];
    }
    __syncthreads();

    int wave = threadIdx.x >> 5;
    int lane = threadIdx.x & 31;
    int tile = blockIdx.x * 8 + wave;
    if (tile >= nTiles) return;          // wave-uniform: EXEC stays all-ones for WMMA

    int half = lane >> 4;                // K sub-pair select (lanes 16-31 hold K=2,3)
    int lm   = lane & 15;                // M (for A) / N (for B,C,D) within tile

    const float* rowp = H + (size_t)(tile * 16 + lm) * NIN + 2 * half;

    v8f acc[4][2] = {};                  // 4 matrices x 2 N-tiles, f32 16x16 accum

    #pragma unroll 4
    for (int kc = 0; kc < 24; ++kc) {
        // A fragment 16x4: lane half picks K={0,1} or K={2,3} of this chunk
        v2f a;
        a.x = rowp[4 * kc + 0];
        a.y = rowp[4 * kc + 1];
        const float* wrow = lw + (4 * kc + 2 * half) * NOUT + lm;
        #pragma unroll
        for (int m = 0; m < 4; ++m) {
            #pragma unroll
            for (int t = 0; t < 2; ++t) {
                v2f b;
                b.x = wrow[m * 3072 + 16 * t];        // K row k, col 16t+lm
                b.y = wrow[m * 3072 + 32 + 16 * t];   // K row k+1
                acc[m][t] = __builtin_amdgcn_wmma_f32_16x16x4_f32(
                    false, a, false, b, (short)0, acc[m][t], false, false);
            }
        }
    }

    // C/D layout: VGPR r, lanes 0-15 -> row M=r, N=lane; lanes 16-31 -> M=r+8, N=lane-16
    int rowbase = tile * 16 + 8 * half;
    #pragma unroll
    for (int t = 0; t < 2; ++t) {
        int   col   = 16 * t + lm;
        float biasA = Ba[col];
        float biasB = Bb[col];
        float biasC = Bc[col];
        #pragma unroll
        for (int r = 0; r < 8; ++r) {
            size_t row = (size_t)(rowbase + r);
            float h1 = fmaxf(acc[0][t][r] + biasA, 0.0f);
            float vb = acc[1][t][r] + biasB;
            float vc = acc[2][t][r] + biasC;
            float h3 = fmaxf(vb * vc, 0.0f);
            Hn[row * NIN + col]       = h1;
            Hn[row * NIN + 64 + col]  = h3;
            pre[row * NOUT + col]     = acc[3][t][r];
        }
    }
}

// ---------------------------------------------------------------------------
// Aggregation as a gather: one wave per node, lane = channel.
// h2 = relu(sum_{e: dst=node} pre[src_e] + conv_b)  ->  H[:,32:64].
// Coalesced 128B reads of L2-resident pre; zero atomics; finalize fused.
// ---------------------------------------------------------------------------
__global__ void __launch_bounds__(256)
gather_agg_kernel(const int* __restrict__ cnt, const int* __restrict__ eidx,
                  const float* __restrict__ pre, const float* __restrict__ convb,
                  float* __restrict__ H)
{
    int node = (blockIdx.x * 256 + threadIdx.x) >> 5;
    if (node >= N_NODES) return;
    int lane = threadIdx.x & 31;
    int deg  = cnt[node];
    deg = deg < CAP ? deg : CAP;
    const int* ep = eidx + (size_t)node * CAP;
    float s = 0.0f;
    int i = 0;
    for (; i + 4 <= deg; i += 4) {
        int s0 = ep[i], s1 = ep[i + 1], s2 = ep[i + 2], s3 = ep[i + 3];
        s += pre[(size_t)s0 * NOUT + lane];
        s += pre[(size_t)s1 * NOUT + lane];
        s += pre[(size_t)s2 * NOUT + lane];
        s += pre[(size_t)s3 * NOUT + lane];
    }
    for (; i < deg; ++i) s += pre[(size_t)ep[i] * NOUT + lane];
    H[(size_t)node * NIN + 32 + lane] = fmaxf(s + convb[lane], 0.0f);
}

__global__ void __launch_bounds__(256)
fill0_kernel(float* __restrict__ p, int n)
{
    int idx = blockIdx.x * 256 + threadIdx.x;
    if (idx < n) p[idx] = 0.0f;
}

// ---------------------------------------------------------------------------
// Pool: batch is sorted, so each graph is a contiguous node range.
// One block per graph: binary-search the range, reduce 96 channels in registers.
// ---------------------------------------------------------------------------
__global__ void __launch_bounds__(128)
pool_kernel(const float* __restrict__ H, const int* __restrict__ batch,
            float* __restrict__ pooled)
{
    __shared__ int srange[2];
    int g = blockIdx.x;
    if (threadIdx.x < 2) {
        int target = g + threadIdx.x;       // lower_bound(batch, target)
        int lo = 0, hi = N_NODES;
        while (lo < hi) {
            int m = (lo + hi) >> 1;
            if (batch[m] < target) lo = m + 1; else hi = m;
        }
        srange[threadIdx.x] = lo;
    }
    __syncthreads();
    int lo = srange[0], hi = srange[1];
    int c = threadIdx.x;
    if (c >= NIN) return;
    float s = 0.0f;
    for (int n = lo; n < hi; ++n) s += H[(size_t)n * NIN + c];
    pooled[(size_t)g * NIN + c] = s;
}

// out[g] = (pooled[g] @ fc1_W + fc1_b) @ fc2_W + fc2_b   — 64x96x32x1, trivial
__global__ void
readout_kernel(const float* __restrict__ pooled,
               const float* __restrict__ W1, const float* __restrict__ b1,
               const float* __restrict__ W2, const float* __restrict__ b2,
               float* __restrict__ out)
{
    int g = threadIdx.x;
    if (g >= N_GRAPHS) return;
    const float* pg = pooled + (size_t)g * NIN;
    float s = b2[0];
    for (int j = 0; j < 32; ++j) {
        float v = b1[j];
        for (int k = 0; k < NIN; ++k) v = fmaf(pg[k], W1[k * 32 + j], v);
        s = fmaf(v, W2[j], s);
    }
    out[g] = s;
}

// ---------------------------------------------------------------------------
extern "C" void kernel_launch(void* const* d_in, const int* in_sizes, int n_in,
                              void* d_out, int out_size, void* d_ws, size_t ws_size,
                              hipStream_t stream) {
    const float* x       = (const float*)d_in[0];
    const int*   ei      = (const int*)  d_in[1];
    const int*   batch   = (const int*)  d_in[2];
    const float* conv1_W = (const float*)d_in[3];
    const float* conv1_b = (const float*)d_in[4];
    const float* fc11_W  = (const float*)d_in[5];
    const float* fc11_b  = (const float*)d_in[6];
    const float* fc12_W  = (const float*)d_in[7];
    const float* fc12_b  = (const float*)d_in[8];
    const float* fc13_W  = (const float*)d_in[9];
    const float* fc13_b  = (const float*)d_in[10];
    const float* conv_W  = (const float*)d_in[11];
    const float* conv_b  = (const float*)d_in[12];
    const float* fcA_W   = (const float*)d_in[13];
    const float* fcA_b   = (const float*)d_in[14];
    const float* fcB_W   = (const float*)d_in[15];
    const float* fcB_b   = (const float*)d_in[16];
    const float* fcC_W   = (const float*)d_in[17];
    const float* fcC_b   = (const float*)d_in[18];
    const float* fc1_W   = (const float*)d_in[19];
    const float* fc1_b   = (const float*)d_in[20];
    const float* fc2_W   = (const float*)d_in[21];
    const float* fc2_b   = (const float*)d_in[22];
    float* out = (float*)d_out;

    char*  ws     = (char*)d_ws;
    float* H      = (float*)(ws);                 // 50000*96 f32   = 19.2 MB
    float* pre    = (float*)(ws + 19200000);      // 50000*32 f32   =  6.4 MB
    int*   cnt    = (int*)  (ws + 25600000);      // 50000 i32      =  0.2 MB
    int*   eidx   = (int*)  (ws + 25800000);      // 50000*64 i32   = 12.8 MB
    float* pooled = (float*)(ws + 38600000);      // 64*96 f32

    const int* src = ei;
    const int* dst = ei + N_EDGES;

    const int nNodeCh = N_NODES * NOUT;                  // 1.6 M
    const int gNodeCh = (nNodeCh + 255) / 256;           // 6250
    const int gGemm   = (NTILES + 7) / 8;                // 391 blocks of 8 waves
    const int gEdges  = (N_EDGES + 255) / 256;           // 3125

    // ---- build in-edge buckets once (graph is fixed across all 5 blocks) ----
    fill0_kernel<<<(N_NODES + 255) / 256, 256, 0, stream>>>((float*)cnt, N_NODES);
    build_csr_kernel<<<gEdges, 256, 0, stream>>>(src, dst, cnt, eidx);

    // ---- block 1 (in = 2) ----
    block1_kernel<<<gNodeCh, 256, 0, stream>>>(x, conv1_W,
                                               fc11_W, fc11_b, fc12_W, fc12_b,
                                               fc13_W, fc13_b, H, pre);
    gather_agg_kernel<<<gNodeCh, 256, 0, stream>>>(cnt, eidx, pre, conv1_b, H);

    // ---- blocks 2..5 (in = 96), WMMA GEMMs ----
    for (int i = 0; i < 4; ++i) {
        gemm96_kernel<<<gGemm, 256, 0, stream>>>(
            H,
            fcA_W + i * NIN * NOUT, fcA_b + i * NOUT,
            fcB_W + i * NIN * NOUT, fcB_b + i * NOUT,
            fcC_W + i * NIN * NOUT, fcC_b + i * NOUT,
            conv_W + i * NIN * NOUT,
            H, pre, NTILES);
        gather_agg_kernel<<<gNodeCh, 256, 0, stream>>>(cnt, eidx, pre,
                                                       conv_b + i * NOUT, H);
    }

    // ---- pool + readout (no atomics; batch is sorted) ----
    pool_kernel<<<N_GRAPHS, 128, 0, stream>>>(H, batch, pooled);
    readout_kernel<<<1, 64, 0, stream>>>(pooled, fc1_W, fc1_b, fc2_W, fc2_b, out);
}